// ScaleDotProductAttention_1546188227176
// MI455X (gfx1250) — compile-verified
//
#include <hip/hip_runtime.h>

// MI455X (gfx1250) fused attention: out = softmax(QK^T/64) V, also emits attn.
// fp32 WMMA (V_WMMA_F32_16X16X4_F32), wave32, one 16-row query tile per wave.

typedef float v2f __attribute__((ext_vector_type(2)));
typedef float v8f __attribute__((ext_vector_type(8)));

static constexpr int S_LEN = 2048;
static constexpr int D_DIM = 64;
static constexpr int BH    = 24;          // B*H = 2*12
static constexpr int WPB   = 4;           // waves per block
static constexpr int ROWS_PER_BLOCK = WPB * 16;

__global__ __launch_bounds__(WPB * 32)
void attn_fused_kernel(const float* __restrict__ Q,
                       const float* __restrict__ Km,
                       const float* __restrict__ Vm,
                       float* __restrict__ Out,
                       float* __restrict__ Attn)
{
    const int tid  = threadIdx.x;
    const int lane = tid & 31;
    const int wave = tid >> 5;
    const int m    = lane & 15;   // M/N index of A/B fragments, col index of C
    const int kh   = lane >> 4;   // selects K pair {0,1} vs {2,3} in fragments

    const int bh = blockIdx.y;
    const int q0 = blockIdx.x * ROWS_PER_BLOCK + wave * 16;

    const float* Qb = Q  + (size_t)bh * S_LEN * D_DIM;
    const float* Kb = Km + (size_t)bh * S_LEN * D_DIM;
    const float* Vb = Vm + (size_t)bh * S_LEN * D_DIM;
    float* Ob = Out  + (size_t)bh * S_LEN * D_DIM;
    float* Ab = Attn + (size_t)bh * (size_t)S_LEN * S_LEN;

    __shared__ float s_m [WPB][16];
    __shared__ float s_il[WPB][16];

    // ---- Preload Q tile (16x64) as 16 A-fragments (16x4 f32, 2 VGPRs each) ----
    v2f qa[16];
#pragma unroll
    for (int f = 0; f < 16; ++f) {
        const float* p = Qb + (size_t)(q0 + m) * D_DIM + 4 * f + 2 * kh;
        qa[f] = *(const v2f*)p;
    }

    float run_m[8], run_l[8];
#pragma unroll
    for (int r = 0; r < 8; ++r) { run_m[r] = -3.0e38f; run_l[r] = 0.0f; }

    const float scale = 1.0f / 64.0f;   // reference divides by d_tensor, not sqrt

    // ============ Phase A: scores = QK^T/64 -> raw to Attn buffer, ============
    // ============ online row max / sum-exp tracked per wave         ============
    for (int j0 = 0; j0 < S_LEN; j0 += 16) {
        // K tile 16x64 as 16 B-fragments (4x16): B[k][n] = K[j0+n][d0+k]
        v2f kb[16];
#pragma unroll
        for (int f = 0; f < 16; ++f) {
            const float* p = Kb + (size_t)(j0 + m) * D_DIM + 4 * f + 2 * kh;
            kb[f] = *(const v2f*)p;
        }
        v8f c = {};
#pragma unroll
        for (int f = 0; f < 16; ++f) {
            c = __builtin_amdgcn_wmma_f32_16x16x4_f32(false, qa[f], false, kb[f],
                                                      (short)0, c, false, false);
        }
#pragma unroll
        for (int r = 0; r < 8; ++r) {
            float cc = c[r] * scale;
            // write raw score (overwritten with softmax value in Phase B)
            Ab[(size_t)(q0 + r + 8 * kh) * S_LEN + (j0 + m)] = cc;

            // row max across the 16 lanes holding this row
            float tmax = cc;
            tmax = fmaxf(tmax, __shfl_xor(tmax, 1));
            tmax = fmaxf(tmax, __shfl_xor(tmax, 2));
            tmax = fmaxf(tmax, __shfl_xor(tmax, 4));
            tmax = fmaxf(tmax, __shfl_xor(tmax, 8));
            float mn = fmaxf(run_m[r], tmax);

            float e = __expf(cc - mn);
            e += __shfl_xor(e, 1);
            e += __shfl_xor(e, 2);
            e += __shfl_xor(e, 4);
            e += __shfl_xor(e, 8);

            run_l[r] = run_l[r] * __expf(run_m[r] - mn) + e;
            run_m[r] = mn;
        }
    }

    // ---- publish per-row stats (max, 1/sum) to this wave's LDS slot ----
    if (m == 0) {                 // lanes 0 and 16: rows 0-7 and 8-15
#pragma unroll
        for (int r = 0; r < 8; ++r) {
            s_m [wave][8 * kh + r] = run_m[r];
            s_il[wave][8 * kh + r] = 1.0f / run_l[r];
        }
    }
    // (same-wave LDS RAW: compiler inserts s_wait_dscnt; no block barrier needed)
    float row_m  = s_m [wave][m];
    float row_il = s_il[wave][m];

    // make Phase A raw-score stores visible before the re-read below
    asm volatile("s_wait_storecnt 0" ::: "memory");

    // ============ Phase B: p = exp(s-m)/l -> Attn; O^T += V^T * P^T ============
    v8f o[4] = {v8f{}, v8f{}, v8f{}, v8f{}};

    for (int j0 = 0; j0 < S_LEN; j0 += 16) {
        // P fragments in B-operand layout: lane n=m holds its query row,
        // pb[f] = { P[m][4f+2kh], P[m][4f+2kh+1] }   (reads hit L2)
        v2f pb[4];
#pragma unroll
        for (int f = 0; f < 4; ++f) {
            float* p = Ab + (size_t)(q0 + m) * S_LEN + j0 + 4 * f + 2 * kh;
            v2f raw = *(const v2f*)p;
            pb[f][0] = __expf(raw[0] - row_m) * row_il;
            pb[f][1] = __expf(raw[1] - row_m) * row_il;
            *(v2f*)p = pb[f];     // final normalized attention weights
        }
        // O^T[d][mq] += sum_k V[k][d] * P[mq][k]
#pragma unroll
        for (int f = 0; f < 4; ++f) {
            const int k0 = j0 + 4 * f + 2 * kh;
#pragma unroll
            for (int dt = 0; dt < 4; ++dt) {
                v2f va;  // A-fragment of V^T: lane -> d, pair -> k
                va[0] = Vb[(size_t)k0       * D_DIM + 16 * dt + m];
                va[1] = Vb[(size_t)(k0 + 1) * D_DIM + 16 * dt + m];
                o[dt] = __builtin_amdgcn_wmma_f32_16x16x4_f32(
                            false, va, false, pb[f], (short)0, o[dt], false, false);
            }
        }
    }

    // ---- store O (o[dt] holds O^T tile: row=d_local, col=query row) ----
#pragma unroll
    for (int dt = 0; dt < 4; ++dt) {
#pragma unroll
        for (int r = 0; r < 8; ++r) {
            Ob[(size_t)(q0 + m) * D_DIM + 16 * dt + r + 8 * kh] = o[dt][r];
        }
    }
}

extern "C" void kernel_launch(void* const* d_in, const int* in_sizes, int n_in,
                              void* d_out, int out_size, void* d_ws, size_t ws_size,
                              hipStream_t stream) {
    const float* Q = (const float*)d_in[0];
    const float* K = (const float*)d_in[1];
    const float* V = (const float*)d_in[2];
    float* Out  = (float*)d_out;                                  // [B,H,S,D]
    float* Attn = Out + (size_t)BH * S_LEN * D_DIM;               // [B,H,S,S]

    dim3 grid(S_LEN / ROWS_PER_BLOCK, BH);
    attn_fused_kernel<<<grid, WPB * 32, 0, stream>>>(Q, K, V, Out, Attn);
}